// DenoiseTransform_85916525789588
// MI455X (gfx1250) — compile-verified
//
#include <hip/hip_runtime.h>

// Bilateral blur 7x7, sigma_color=10, sigma_space=21, reflect-101 border,
// L1 color distance.  Input/output: (8,3,512,512) fp32 NCHW.
//
// Strategy (MI455X): compute-bound stencil -> stage halo tile in LDS via
// CDNA5 async global->LDS loads (ASYNCcnt), serve all 49 taps from LDS as
// b128 loads (RGBx interleaved), one v_exp_f32 per tap with all weights
// folded into the exp2 argument in log2 domain.

#define TILE_W 64
#define TILE_H 16
#define HALO   3
#define KS     7
#define PW     (TILE_W + 2 * HALO)   // 70
#define PH     (TILE_H + 2 * HALO)   // 22
#define IMG_H  512
#define IMG_W  512
#define NCH    3
#define NBATCH 8

typedef __attribute__((address_space(1))) int gas_int;
typedef __attribute__((address_space(3))) int las_int;

__device__ __forceinline__ int reflect512(int q) {
  // jnp.pad mode="reflect" (reflect-101): -1 -> 1, 512 -> 510
  q = (q < 0) ? -q : q;
  q = (q >= IMG_W) ? (2 * IMG_W - 2 - q) : q;
  return q;
}

__device__ __forceinline__ void async_g2l_b32(const float* g, float* l) {
#if __has_builtin(__builtin_amdgcn_global_load_async_to_lds_b32)
  __builtin_amdgcn_global_load_async_to_lds_b32((gas_int*)g, (las_int*)l, 0, 0);
#else
  unsigned lds32 = (unsigned)(unsigned long long)l;   // low 32 bits = LDS offset
  asm volatile("global_load_async_to_lds_b32 %0, %1, off"
               :: "v"(lds32), "v"((unsigned long long)g) : "memory");
#endif
}

__device__ __forceinline__ void wait_async0() {
#if __has_builtin(__builtin_amdgcn_s_wait_asynccnt)
  __builtin_amdgcn_s_wait_asynccnt(0);
#else
  asm volatile("s_wait_asynccnt 0" ::: "memory");
#endif
}

__global__ __launch_bounds__(256) void bilateral7_kernel(
    const float* __restrict__ img, float* __restrict__ out) {
  // RGBx interleaved tile: 16B-aligned pixel slots -> ds_load_b128 per tap.
  __shared__ float tile[PH * PW * 4];

  const int tid = threadIdx.x;
  const int bx = blockIdx.x, by = blockIdx.y, b = blockIdx.z;
  const int gx0 = bx * TILE_W - HALO;
  const int gy0 = by * TILE_H - HALO;

  // ---- async fill of halo tile (per-lane gather handles reflect border) ----
  const int NELEM = PH * PW * NCH;  // 4620 -> ~18 async ops / thread (< 63 cnt)
  for (int e = tid; e < NELEM; e += 256) {
    const int c   = e / (PH * PW);
    const int rem = e - c * (PH * PW);
    const int r   = rem / PW;
    const int col = rem - r * PW;
    const int gy = reflect512(gy0 + r);
    const int gx = reflect512(gx0 + col);
    const float* gp = img + ((((size_t)b * NCH + c) * IMG_H + gy) * IMG_W + gx);
    float* lp = &tile[(r * PW + col) * 4 + c];
    async_g2l_b32(gp, lp);
  }
  wait_async0();     // our wave's async LDS writes are visible
  __syncthreads();   // all waves' fills complete

  // Unnormalized spatial Gaussian in log2 domain (normalization cancels in
  // num/den):  lg[i] = -0.5*log2(e)*((i-3)/21)^2
  const float lg[KS] = {-0.01472137797f, -0.00654283465f, -0.00163570866f, 0.0f,
                        -0.00163570866f, -0.00654283465f, -0.01472137797f};
  // color term: (-0.5 / 10^2) * log2(e)
  const float c2 = -0.00721347520f;

  const int px  = tid & (TILE_W - 1);   // column within tile (coalesced)
  const int py0 = (tid >> 6) * 4;       // 4 stacked rows per thread

#pragma unroll
  for (int s = 0; s < 4; ++s) {
    const int py = py0 + s;
    const float4 cpx =
        *(const float4*)&tile[((py + HALO) * PW + (px + HALO)) * 4];
    float n0 = 0.f, n1 = 0.f, n2 = 0.f, den = 0.f;
#pragma unroll
    for (int i = 0; i < KS; ++i) {
#pragma unroll
      for (int j = 0; j < KS; ++j) {
        const float4 p = *(const float4*)&tile[((py + i) * PW + (px + j)) * 4];
        const float d = __builtin_fabsf(p.x - cpx.x) +
                        __builtin_fabsf(p.y - cpx.y) +
                        __builtin_fabsf(p.z - cpx.z);
        // single v_exp_f32 carries spatial weight + color weight
        const float w =
            __builtin_amdgcn_exp2f(__builtin_fmaf(c2, d * d, lg[i] + lg[j]));
        n0 = __builtin_fmaf(w, p.x, n0);
        n1 = __builtin_fmaf(w, p.y, n1);
        n2 = __builtin_fmaf(w, p.z, n2);
        den += w;
      }
    }
    const float inv = 1.0f / den;
    const int gy = by * TILE_H + py;
    const int gx = bx * TILE_W + px;
    const size_t plane = (size_t)IMG_H * IMG_W;
    const size_t o0 = (((size_t)b * NCH + 0) * IMG_H + gy) * IMG_W + gx;
    out[o0]             = n0 * inv;
    out[o0 + plane]     = n1 * inv;
    out[o0 + 2 * plane] = n2 * inv;
  }
}

extern "C" void kernel_launch(void* const* d_in, const int* in_sizes, int n_in,
                              void* d_out, int out_size, void* d_ws,
                              size_t ws_size, hipStream_t stream) {
  (void)in_sizes; (void)n_in; (void)out_size; (void)d_ws; (void)ws_size;
  const float* img = (const float*)d_in[0];
  float* out = (float*)d_out;
  dim3 grid(IMG_W / TILE_W, IMG_H / TILE_H, NBATCH);  // (8, 32, 8)
  bilateral7_kernel<<<grid, dim3(256, 1, 1), 0, stream>>>(img, out);
}